// LeakyIntegrator_11570641896145
// MI455X (gfx1250) — compile-verified
//
#include <hip/hip_runtime.h>

// LIF leaky-integrator scan  v_t = v_{t-1} + (x_t - v_{t-1})/tau, tau = 2.
// Reformulated as a blocked triangular matmul on the CDNA5 matrix pipe:
//   v_tile[16 x 16ch] = Tri16 @ x_tile + lambda^(i+1) * v_carry
// with Tri16[i,j] = (1/tau) * lambda^(i-j) (i>=j), lambda = 1 - 1/tau = 0.5.
// Four chained V_WMMA_F32_16X16X4_F32 per tile (K accumulation 4*4 = 16).
// Memory-bound problem: 268 MB @ 23.3 TB/s => ~11.5 us floor; WMMA shortens
// the serial chain 1024 -> 64 steps and keeps the VALU free.

typedef float v2f __attribute__((ext_vector_type(2)));
typedef float v4f __attribute__((ext_vector_type(4)));
typedef float v8f __attribute__((ext_vector_type(8)));

#define T_STEPS 1024
#define CHANS   (16 * 2048)   // B * N = 32768 channels
#define INV_TAU 0.5f

// lambda = 0.5 exactly, so lambda^p = 2^-p: build the fp32 bit pattern
// directly (exponent field = 127 - p). Exact, no libm needed.
__device__ __forceinline__ float lampow(int p) {
  return __int_as_float((127 - p) << 23);
}

__global__ __launch_bounds__(256) void lif_scan_wmma(const float* __restrict__ x,
                                                     float* __restrict__ out) {
  // One wave32 owns 16 consecutive channels for all T.
  const int wave = blockIdx.x * (blockDim.x >> 5) + (threadIdx.x >> 5);
  const int lane = threadIdx.x & 31;
  const int j    = lane & 15;      // channel-in-group / matrix column (also row m for A)
  const int hi   = lane >> 4;      // lane half selects K (A/B) or row+8 (C/D)
  const long c0  = (long)wave * 16;

  // A operands: constant triangular decay matrix, 16x4 slices, 2 VGPRs each.
  // A[k] vgpr0 = Tri[m][4k+2hi], vgpr1 = Tri[m][4k+2hi+1], m = j.
  // Tri[m][c] = INV_TAU * lambda^(m-c) = 0.5 * 2^-(m-c) = 2^-(m-c+1).
  v2f a[4];
#pragma unroll
  for (int k = 0; k < 4; ++k) {
    const int col0 = 4 * k + 2 * hi;
    const int col1 = col0 + 1;
    a[k].x = (j >= col0) ? lampow(j - col0 + 1) : 0.0f;
    a[k].y = (j >= col1) ? lampow(j - col1 + 1) : 0.0f;
  }
  // C-matrix carry scales: C[r] = lambda^(m+1) * carry, m = r + 8*hi.
  float cscale[8];
#pragma unroll
  for (int r = 0; r < 8; ++r)
    cscale[r] = lampow(r + 8 * hi + 1);

  const float* px = x + c0 + j;
  float* po = out + c0 + j;

  // Double-buffered B operands: bx[k] = x[tb+4k+2hi], by[k] = x[tb+4k+1+2hi].
  v4f bxc, byc;
#pragma unroll
  for (int k = 0; k < 4; ++k) {
    bxc[k] = px[(long)(4 * k + 2 * hi) * CHANS];
    byc[k] = px[(long)(4 * k + 1 + 2 * hi) * CHANS];
  }

  float carry = 0.0f;                 // v at t = tb-1 for this lane's column
  const int NT = T_STEPS / 16;        // 64 sequential tiles

  for (int tt = 0; tt < NT; ++tt) {
    const int tb = tt * 16;

    // Prefetch next tile while this tile's WMMAs run.
    v4f bxn, byn;
    if (tt + 1 < NT) {
      const int tb2 = tb + 16;
#pragma unroll
      for (int k = 0; k < 4; ++k) {
        bxn[k] = px[(long)(tb2 + 4 * k + 2 * hi) * CHANS];
        byn[k] = px[(long)(tb2 + 4 * k + 1 + 2 * hi) * CHANS];
      }
    } else {
      bxn = (v4f)0.0f; byn = (v4f)0.0f;
    }

    // C input carries the decayed state: C[i][j] = lambda^(i+1) * v_carry[j].
    v8f d;
#pragma unroll
    for (int r = 0; r < 8; ++r) d[r] = cscale[r] * carry;

    // D = Tri16 @ x_tile + C via 4 chained 16x16x4 f32 WMMAs.
#pragma unroll
    for (int k = 0; k < 4; ++k) {
      v2f b; b.x = bxc[k]; b.y = byc[k];
      d = __builtin_amdgcn_wmma_f32_16x16x4_f32(
          /*neg_a=*/false, a[k], /*neg_b=*/false, b,
          /*c_mod=*/(short)0, d, /*reuse_a=*/false, /*reuse_b=*/false);
    }

    // New carry = D row 15 (VGPR7, lanes 16..31) broadcast to both lane halves.
    {
      const float last = d[7];
      const int srcByte = (16 + j) << 2;
      carry = __int_as_float(
          __builtin_amdgcn_ds_bpermute(srcByte, __float_as_int(last)));
    }

    // Store v_seq tile: row m = r + 8*hi, column j.
#pragma unroll
    for (int r = 0; r < 8; ++r)
      po[(long)(tb + r + 8 * hi) * CHANS] = d[r];

    bxc = bxn; byc = byn;
  }
}

extern "C" void kernel_launch(void* const* d_in, const int* in_sizes, int n_in,
                              void* d_out, int out_size, void* d_ws, size_t ws_size,
                              hipStream_t stream) {
  const float* x = (const float*)d_in[0];
  float* out = (float*)d_out;
  // 32768 channels / 16 per wave = 2048 waves; 8 waves (256 threads) per block.
  dim3 block(256);
  dim3 grid(CHANS / 128);
  lif_scan_wmma<<<grid, block, 0, stream>>>(x, out);
}